// InvariantMapping_46308337385516
// MI455X (gfx1250) — compile-verified
//
#include <hip/hip_runtime.h>
#include <hip/hip_bf16.h>
#include <math.h>

// Problem constants (from reference setup_inputs):
// fx, fy : (B=16, C=128, D=3, N=4096) float32, topk=4 -> k = N/topk = 1024
constexpr int kB = 16;
constexpr int kC = 128;
constexpr int kD = 3;
constexpr int kN = 4096;
constexpr int kK = 1024;            // N / topk
constexpr float kEps = 1e-6f;

constexpr int kCK = 8;              // c-chunk staged per step (2 WMMA K-steps)
constexpr int kNB = 32;             // n's per block (one 128B line per plane)
constexpr int kStageElems = kCK * kD * kNB;  // 768 floats per array per chunk

typedef float v2f __attribute__((ext_vector_type(2)));
typedef float v8f __attribute__((ext_vector_type(8)));

#if __has_builtin(__builtin_amdgcn_global_load_async_to_lds_b32) && \
    __has_builtin(__builtin_amdgcn_s_wait_asynccnt)
#define USE_ASYNC_LDS 1
typedef __attribute__((address_space(1))) int gas_i32;   // global int
typedef __attribute__((address_space(3))) int las_i32;   // LDS int
#endif

// Stage one (fx,fy) c-chunk for 32 consecutive n into LDS, fully coalesced:
// element e = (cc*3 + d)*32 + j  ->  global offset (c0*3 + e>>5)*kN + nbase + j.
__device__ __forceinline__ void stage_chunk(const float* __restrict__ gx,
                                            const float* __restrict__ gy,
                                            float* __restrict__ sx,
                                            float* __restrict__ sy,
                                            int c0, int nbase, int t) {
#pragma unroll
  for (int i = 0; i < kStageElems / 256; ++i) {
    const int e = t + i * 256;
    const size_t goff = (size_t)(c0 * kD + (e >> 5)) * kN + nbase + (e & 31);
#ifdef USE_ASYNC_LDS
    __builtin_amdgcn_global_load_async_to_lds_b32(
        (gas_i32*)(gx + goff), (las_i32*)(sx + e), 0, 0);
    __builtin_amdgcn_global_load_async_to_lds_b32(
        (gas_i32*)(gy + goff), (las_i32*)(sy + e), 0, 0);
#else
    sx[e] = gx[goff];
    sy[e] = gy[goff];
#endif
  }
}

__device__ __forceinline__ void stage_commit() {
#ifdef USE_ASYNC_LDS
  __builtin_amdgcn_s_wait_asynccnt(0);
#endif
  __syncthreads();
}

// ---------------------------------------------------------------------------
// Phase 1: per-(b,n) score via f32 WMMA with LDS-staged, coalesced loads.
// A 256-thread block owns 32 consecutive n; each of its 8 waves owns 4 n.
// WMMA 16x16 tile rows/cols are (nl*4 + d), with d==3 a constant-ones row,
// so chaining C += A(16x4)*B(4x16) over c yields, per diagonal 4x4 block:
//   G[d][d'] = sum_c fx[c,d,n]*fy[c,d',n],  sum_c fx (col 3), sum_c fy (row 3)
// from which  s = (mx^T G my) / ((|mx|+eps)(|my|+eps)),  mx = mean_c fx.
// ---------------------------------------------------------------------------
__global__ void __launch_bounds__(256)
score_wmma_kernel(const float* __restrict__ fx, const float* __restrict__ fy,
                  float* __restrict__ s_out) {
  __shared__ float sgx[2][kStageElems];   // 2 x 3KB
  __shared__ float sgy[2][kStageElems];   // 2 x 3KB
  __shared__ float cmat[8 * 256];         // 8 waves x (8 VGPRs x 32 lanes)

  const int t     = threadIdx.x;
  const int lane  = t & 31;
  const int w     = t >> 5;                       // wave in block
  const int b     = blockIdx.x >> 7;              // 128 n-groups per batch
  const int nbase = (blockIdx.x & 127) << 5;      // 32 n per block
  const int n0    = nbase + (w << 2);             // 4 n per wave

  // Fragment decode (A 16x4 f32: lanes 0-15 K=vgpr, lanes 16-31 K=vgpr+2;
  // B 4x16 mirrored).  Row/col M = 4*nl + d; d==3 is the ones row.
  const int m    = lane & 15;
  const int kh2  = (lane >> 4) << 1;              // 0 or 2
  const int nl   = m >> 2;
  const int d    = m & 3;
  const bool real = (d < 3);
  const int dd   = real ? d : 0;                  // safe LDS address
  const int wj   = (w << 2) + nl;                 // n index within the 32

  const float* gx = fx + (size_t)b * kC * kD * kN;
  const float* gy = fy + (size_t)b * kC * kD * kN;

  stage_chunk(gx, gy, sgx[0], sgy[0], 0, nbase, t);
  stage_commit();

  v8f acc = {};
  for (int ch = 0; ch < kC / kCK; ++ch) {
    const int cur = ch & 1;
    if (ch + 1 < kC / kCK)
      stage_chunk(gx, gy, sgx[cur ^ 1], sgy[cur ^ 1], (ch + 1) * kCK, nbase, t);

    const float* sA = sgx[cur];
    const float* sB = sgy[cur];
#pragma unroll
    for (int s = 0; s < kCK / 4; ++s) {
      const int cc = s * 4 + kh2;                 // local c for vgpr 0
      // Unconditional LDS reads (safe dd), then branchless ones-row select.
      float a0 = sA[((cc + 0) * kD + dd) * kNB + wj];
      float a1 = sA[((cc + 1) * kD + dd) * kNB + wj];
      float b0 = sB[((cc + 0) * kD + dd) * kNB + wj];
      float b1 = sB[((cc + 1) * kD + dd) * kNB + wj];
      a0 = real ? a0 : 1.0f;
      a1 = real ? a1 : 1.0f;
      b0 = real ? b0 : 1.0f;
      b1 = real ? b1 : 1.0f;
      v2f av = {a0, a1};
      v2f bv = {b0, b1};
      // 8 args: (neg_a, A, neg_b, B, c_mod, C, reuse_a, reuse_b)
      acc = __builtin_amdgcn_wmma_f32_16x16x4_f32(
          false, av, false, bv, (short)0, acc, false, false);
    }
    stage_commit();   // next buffer landed; current buffer free to overwrite
  }

  // Spill accumulator so designated lanes can read the diagonal 4x4 blocks.
  float* my_c = cmat + w * 256;
#pragma unroll
  for (int v = 0; v < 8; ++v) my_c[v * 32 + lane] = acc[v];
  __syncthreads();

  if (lane < 4) {
    const int base = lane * 4;                    // this lane owns n0+lane
    auto val = [&](int M, int Nc) -> float {
      // C/D layout: vgpr = M%8, lane = Nc + 16*(M/8)
      return my_c[(M & 7) * 32 + (Nc + ((M >> 3) << 4))];
    };
    float mx[3], my[3], G[3][3];
#pragma unroll
    for (int i = 0; i < 3; ++i) {
      mx[i] = val(base + i, base + 3) * (1.0f / (float)kC);
      my[i] = val(base + 3, base + i) * (1.0f / (float)kC);
#pragma unroll
      for (int j = 0; j < 3; ++j) G[i][j] = val(base + i, base + j);
    }
    const float nx = sqrtf(mx[0]*mx[0] + mx[1]*mx[1] + mx[2]*mx[2]) + kEps;
    const float ny = sqrtf(my[0]*my[0] + my[1]*my[1] + my[2]*my[2]) + kEps;
    float num = 0.0f;
#pragma unroll
    for (int i = 0; i < 3; ++i)
#pragma unroll
      for (int j = 0; j < 3; ++j) num += mx[i] * G[i][j] * my[j];
    s_out[(size_t)b * kN + n0 + lane] = num / (nx * ny);
  }
}

// ---------------------------------------------------------------------------
// Phase 2: exact ordered top-k per batch by rank counting (matches
// jax.lax.top_k incl. tie-break toward lower index; softmax skipped since it
// is order-preserving).  rank(i) = #{j : s[j] > s[i] or (s[j]==s[i], j<i)}.
// ---------------------------------------------------------------------------
__global__ void __launch_bounds__(256)
topk_kernel(const float* __restrict__ s, int* __restrict__ idx) {
  __shared__ float vals[kN];
  const int b = blockIdx.x;
  const float* sb = s + (size_t)b * kN;
  for (int i = threadIdx.x; i < kN; i += blockDim.x) vals[i] = sb[i];
  __syncthreads();
  for (int i = threadIdx.x; i < kN; i += blockDim.x) {
    const float v = vals[i];
    int rank = 0;
    for (int j = 0; j < kN; ++j) {
      const float u = vals[j];
      rank += (u > v) || (u == v && j < i);
    }
    if (rank < kK) idx[b * kK + rank] = i;
  }
}

// ---------------------------------------------------------------------------
// Phase 3: rank-ordered gather.  Output layout: fx_sel (B,C,D,kK) then fy_sel.
// Writes fully coalesced over r; reads mostly hit L2 (192 MB).
// ---------------------------------------------------------------------------
__global__ void __launch_bounds__(256)
gather_kernel(const float* __restrict__ fx, const float* __restrict__ fy,
              const int* __restrict__ idx, float* __restrict__ out) {
  const size_t OUT1 = (size_t)kB * kC * kD * kK;
  const size_t gid = (size_t)blockIdx.x * blockDim.x + threadIdx.x;
  if (gid >= OUT1) return;
  const int r = (int)(gid & (kK - 1));
  size_t tt = gid >> 10;                    // / kK
  const int d = (int)(tt % kD);  tt /= kD;
  const int c = (int)(tt % kC);
  const int b = (int)(tt / kC);
  const int n = idx[b * kK + r];
  const size_t src = (((size_t)b * kC + c) * kD + d) * kN + n;
  out[gid]        = fx[src];
  out[OUT1 + gid] = fy[src];
}

extern "C" void kernel_launch(void* const* d_in, const int* in_sizes, int n_in,
                              void* d_out, int out_size, void* d_ws, size_t ws_size,
                              hipStream_t stream) {
  const float* fx = (const float*)d_in[0];
  const float* fy = (const float*)d_in[1];
  // d_in[2] is topk==4 (fixed by setup_inputs); kK = kN/4 baked in.

  float* s_buf  = (float*)d_ws;                                        // 256 KB
  int*   idxbuf = (int*)((char*)d_ws + (size_t)kB * kN * sizeof(float)); // 64 KB

  // Phase 1: one block per 32 n (8 waves x 4 n), 2048 blocks.
  score_wmma_kernel<<<kB * (kN / kNB), 256, 0, stream>>>(fx, fy, s_buf);

  // Phase 2: one block per batch.
  topk_kernel<<<kB, 256, 0, stream>>>(s_buf, idxbuf);

  // Phase 3: one thread per selected (b,c,d,r) element.
  const size_t OUT1 = (size_t)kB * kC * kD * kK;
  gather_kernel<<<(unsigned)((OUT1 + 255) / 256), 256, 0, stream>>>(
      fx, fy, idxbuf, (float*)d_out);
}